// NPool_88656714924905
// MI455X (gfx1250) — compile-verified
//
#include <hip/hip_runtime.h>
#include <hip/hip_bf16.h>

typedef float v2f __attribute__((ext_vector_type(2)));
typedef float v8f __attribute__((ext_vector_type(8)));

#define NEG_SLOPE 0.2f
#define LAMB 0.7f
#define SORT_LOCAL 2048   // elements per LDS sort chunk (16 KB of u64)

// ---- ordered-uint encoding of float for atomic max / sort keys ----
__device__ __forceinline__ unsigned int enc_f32(float f) {
    unsigned int u = __float_as_uint(f);
    return (u & 0x80000000u) ? ~u : (u | 0x80000000u);
}
__device__ __forceinline__ float dec_f32(unsigned int e) {
    unsigned int u = (e & 0x80000000u) ? (e ^ 0x80000000u) : ~e;
    return __uint_as_float(u);
}

// ---- init workspace arrays ----
__global__ void k_init(float* __restrict__ xagg, unsigned int* __restrict__ menc,
                       float* __restrict__ den, float* __restrict__ ssum,
                       float* __restrict__ agg, int* __restrict__ mapping,
                       int N, int C) {
    long total = (long)N * C;
    for (long i = blockIdx.x * (long)blockDim.x + threadIdx.x; i < total;
         i += (long)gridDim.x * blockDim.x) {
        xagg[i] = 0.0f;
        if (i < N) {
            menc[i] = 0u; den[i] = 0.0f; ssum[i] = 0.0f; agg[i] = 0.0f;
            mapping[i] = -1;
        }
    }
}

// ---- skinny GEMV via WMMA f32 16x16x4: p_src = x@w_att[:C], p_dst = x@w_att[C:]
// One wave handles 16 rows; B-matrix column 0 = w_src, column 1 = w_dst, rest 0.
__global__ void k_gemv_att(const float* __restrict__ x, const float* __restrict__ w_att,
                           float* __restrict__ p_src, float* __restrict__ p_dst,
                           int N, int C) {
    int wave = blockIdx.x * (blockDim.x >> 5) + (threadIdx.x >> 5);
    int lane = threadIdx.x & 31;
    int rowm = lane & 15;       // M index (A) / N index (B,C,D)
    int half = lane >> 4;       // selects K pair
    int base = wave * 16;
    if (base >= N) return;      // uniform per wave
    int row = base + rowm; if (row >= N) row = N - 1;
    const float* xr = x + (long)row * C + half * 2;
    float msk0 = (rowm == 0) ? 1.0f : 0.0f;
    float msk1 = (rowm == 1) ? 1.0f : 0.0f;
    v8f acc = {0.f,0.f,0.f,0.f,0.f,0.f,0.f,0.f};
    for (int k0 = 0; k0 < C; k0 += 4) {
        v2f a; a.x = xr[k0]; a.y = xr[k0 + 1];
        int kb = k0 + half * 2;
        v2f b;
        b.x = msk0 * w_att[kb]     + msk1 * w_att[C + kb];
        b.y = msk0 * w_att[kb + 1] + msk1 * w_att[C + kb + 1];
        acc = __builtin_amdgcn_wmma_f32_16x16x4_f32(false, a, false, b,
                                                    (short)0, acc, false, false);
    }
    if (rowm < 2) {
        float* outp = (rowm == 0) ? p_src : p_dst;
        for (int v = 0; v < 8; ++v) {
            int r = base + v + 8 * half;
            if (r < N) outp[r] = acc[v];
        }
    }
}

// ---- LEConv GEMV via WMMA: cols = {w1 (+b1), w2, w3} on x_agg ----
__global__ void k_gemv_le(const float* __restrict__ xagg,
                          const float* __restrict__ w1, const float* __restrict__ w2,
                          const float* __restrict__ w3, const float* __restrict__ b1,
                          float* __restrict__ aout, float* __restrict__ bout,
                          float* __restrict__ cout, int N, int C) {
    int wave = blockIdx.x * (blockDim.x >> 5) + (threadIdx.x >> 5);
    int lane = threadIdx.x & 31;
    int rowm = lane & 15;
    int half = lane >> 4;
    int base = wave * 16;
    if (base >= N) return;
    int row = base + rowm; if (row >= N) row = N - 1;
    const float* xr = xagg + (long)row * C + half * 2;
    const float* wsel = (rowm == 1) ? w2 : ((rowm == 2) ? w3 : w1);
    float msk = (rowm < 3) ? 1.0f : 0.0f;
    v8f acc = {0.f,0.f,0.f,0.f,0.f,0.f,0.f,0.f};
    for (int k0 = 0; k0 < C; k0 += 4) {
        v2f a; a.x = xr[k0]; a.y = xr[k0 + 1];
        int kb = k0 + half * 2;
        v2f b; b.x = msk * wsel[kb]; b.y = msk * wsel[kb + 1];
        acc = __builtin_amdgcn_wmma_f32_16x16x4_f32(false, a, false, b,
                                                    (short)0, acc, false, false);
    }
    if (rowm < 3) {
        float bias = (rowm == 0) ? b1[0] : 0.0f;
        float* outp = (rowm == 0) ? aout : ((rowm == 1) ? bout : cout);
        for (int v = 0; v < 8; ++v) {
            int r = base + v + 8 * half;
            if (r < N) outp[r] = acc[v] + bias;
        }
    }
}

// ---- edge scores + segment max (ordered-uint atomicMax) ----
__global__ void k_score(const int* __restrict__ esrc, const int* __restrict__ edst,
                        const float* __restrict__ p_src, const float* __restrict__ p_dst,
                        const float* __restrict__ b_att,
                        float* __restrict__ score, unsigned int* __restrict__ menc,
                        int E, int M) {
    int i = blockIdx.x * blockDim.x + threadIdx.x;
    if (i >= M) return;
    int s = (i < E) ? esrc[i] : (i - E);
    int d = (i < E) ? edst[i] : (i - E);
    float v = p_src[s] + p_dst[d] + b_att[0];
    v = (v > 0.0f) ? v : NEG_SLOPE * v;
    score[i] = v;
    atomicMax(&menc[d], enc_f32(v));
}

// ---- exp, denominator, and score_sum (self-loops excluded from score_sum) ----
__global__ void k_exp(const int* __restrict__ edst, const float* __restrict__ score,
                      const unsigned int* __restrict__ menc,
                      float* __restrict__ exbuf, float* __restrict__ den,
                      float* __restrict__ ssum, int E, int M) {
    int i = blockIdx.x * blockDim.x + threadIdx.x;
    if (i >= M) return;
    int d = (i < E) ? edst[i] : (i - E);
    float sc = score[i];
    float ex = expf(sc - dec_f32(menc[d]));
    exbuf[i] = ex;
    atomicAdd(&den[d], ex);
    if (i < E) atomicAdd(&ssum[d], sc);
}

// ---- heavy scatter: x_agg[dst] += alpha * x[src]; one wave per edge, float4/lane
__global__ void k_agg(const int* __restrict__ esrc, const int* __restrict__ edst,
                      const float* __restrict__ x, const float* __restrict__ exbuf,
                      const float* __restrict__ den, float* __restrict__ xagg,
                      int E, int M, int C) {
    long t = blockIdx.x * (long)blockDim.x + threadIdx.x;
    long edge = t >> 5;
    if (edge >= M) return;
    int lane = (int)(t & 31);
    int s = (edge < E) ? esrc[edge] : (int)(edge - E);
    int d = (edge < E) ? edst[edge] : (int)(edge - E);
    float alpha = exbuf[edge] / den[d];
    float4 v = ((const float4*)(x + (long)s * C))[lane];
    float* dp = xagg + (long)d * C + lane * 4;
    atomicAdd(dp + 0, v.x * alpha);
    atomicAdd(dp + 1, v.y * alpha);
    atomicAdd(dp + 2, v.z * alpha);
    atomicAdd(dp + 3, v.w * alpha);
}

// ---- LEConv aggregation: agg[dst] += a[src] - b[dst] over all M edges ----
__global__ void k_leagg(const int* __restrict__ esrc, const int* __restrict__ edst,
                        const float* __restrict__ a, const float* __restrict__ b,
                        float* __restrict__ agg, int E, int M) {
    int i = blockIdx.x * blockDim.x + threadIdx.x;
    if (i >= M) return;
    int s = (i < E) ? esrc[i] : (i - E);
    int d = (i < E) ? edst[i] : (i - E);
    atomicAdd(&agg[d], a[s] - b[d]);
}

// ---- fitness + final score -> sort keys (value desc, index asc on ties) ----
__global__ void k_fitness(const float* __restrict__ agg, const float* __restrict__ xw3,
                          const float* __restrict__ b3, const float* __restrict__ ssum,
                          unsigned long long* __restrict__ keys, int N, int PAD) {
    int i = blockIdx.x * blockDim.x + threadIdx.x;
    if (i >= PAD) return;
    if (i < N) {
        float g = agg[i] + xw3[i] + b3[0];
        float fit = 1.0f / (1.0f + expf(-g));
        float fs = fit - LAMB * ssum[i];
        keys[i] = ((unsigned long long)enc_f32(fs) << 32) |
                  (unsigned long long)(0xFFFFFFFFu - (unsigned int)i);
    } else {
        keys[i] = 0ull;  // smaller than any encoded real value
    }
}

// ---- global bitonic pass (descending), for j >= SORT_LOCAL ----
__global__ void k_bitonic(unsigned long long* __restrict__ keys, int j, int k, int PAD) {
    int i = blockIdx.x * blockDim.x + threadIdx.x;
    if (i >= PAD) return;
    int ixj = i ^ j;
    if (ixj > i) {
        unsigned long long a = keys[i], b = keys[ixj];
        bool up = ((i & k) == 0);
        bool sw = up ? (a < b) : (a > b);   // descending sort
        if (sw) { keys[i] = b; keys[ixj] = a; }
    }
}

// ---- LDS presort: fully sort 2048-element chunks (stages k=2..2048) ----
__global__ void __launch_bounds__(SORT_LOCAL / 2)
k_bitonic_presort(unsigned long long* __restrict__ keys) {
    __shared__ unsigned long long lds[SORT_LOCAL];
    const int t = threadIdx.x;                       // 0..1023
    const long base = (long)blockIdx.x * SORT_LOCAL;
    lds[t] = keys[base + t];
    lds[t + SORT_LOCAL / 2] = keys[base + t + SORT_LOCAL / 2];
    __syncthreads();
    for (int k2 = 2; k2 <= SORT_LOCAL; k2 <<= 1) {
        for (int j = k2 >> 1; j > 0; j >>= 1) {
            int i = ((t & ~(j - 1)) << 1) | (t & (j - 1));
            int p = i | j;
            bool up = (((base + i) & (long)k2) == 0);
            unsigned long long a = lds[i], b = lds[p];
            bool sw = up ? (a < b) : (a > b);
            if (sw) { lds[i] = b; lds[p] = a; }
            __syncthreads();
        }
    }
    keys[base + t] = lds[t];
    keys[base + t + SORT_LOCAL / 2] = lds[t + SORT_LOCAL / 2];
}

// ---- LDS merge: for outer stage k2 (> SORT_LOCAL), passes j = 1024..1 ----
__global__ void __launch_bounds__(SORT_LOCAL / 2)
k_bitonic_merge_lds(unsigned long long* __restrict__ keys, int k2) {
    __shared__ unsigned long long lds[SORT_LOCAL];
    const int t = threadIdx.x;
    const long base = (long)blockIdx.x * SORT_LOCAL;
    lds[t] = keys[base + t];
    lds[t + SORT_LOCAL / 2] = keys[base + t + SORT_LOCAL / 2];
    __syncthreads();
    const bool up = ((base & (long)k2) == 0);        // uniform per chunk (k2 > chunk)
    for (int j = SORT_LOCAL >> 1; j > 0; j >>= 1) {
        int i = ((t & ~(j - 1)) << 1) | (t & (j - 1));
        int p = i | j;
        unsigned long long a = lds[i], b = lds[p];
        bool sw = up ? (a < b) : (a > b);
        if (sw) { lds[i] = b; lds[p] = a; }
        __syncthreads();
    }
    keys[base + t] = lds[t];
    keys[base + t + SORT_LOCAL / 2] = lds[t + SORT_LOCAL / 2];
}

// ---- build mapping[perm[k]] = k ----
__global__ void k_map(const unsigned long long* __restrict__ keys,
                      int* __restrict__ mapping, int K) {
    int k = blockIdx.x * blockDim.x + threadIdx.x;
    if (k >= K) return;
    unsigned int perm = 0xFFFFFFFFu - (unsigned int)(keys[k] & 0xFFFFFFFFull);
    mapping[perm] = k;
}

// ---- x_out = x_agg[perm] * vals; also batch_out=0 and perm (as floats) ----
__global__ void k_outx(const unsigned long long* __restrict__ keys,
                       const float* __restrict__ xagg,
                       float* __restrict__ out_x, float* __restrict__ out_batch,
                       float* __restrict__ out_perm, int K, int C) {
    long t = blockIdx.x * (long)blockDim.x + threadIdx.x;
    int k = (int)(t >> 5);
    if (k >= K) return;
    int lane = (int)(t & 31);
    unsigned long long key = keys[k];
    unsigned int perm = 0xFFFFFFFFu - (unsigned int)(key & 0xFFFFFFFFull);
    float val = dec_f32((unsigned int)(key >> 32));
    float4 v = ((const float4*)(xagg + (long)perm * C))[lane];
    float4* o = (float4*)(out_x + (long)k * C) + lane;
    *o = make_float4(v.x * val, v.y * val, v.z * val, v.w * val);
    if (lane == 0) {
        out_batch[k] = 0.0f;
        out_perm[k] = (float)perm;
    }
}

// ---- relabel edges: keep iff both endpoints kept, else -1 (both) ----
__global__ void k_edges(const int* __restrict__ esrc, const int* __restrict__ edst,
                        const int* __restrict__ mapping, float* __restrict__ out_e,
                        int E, int M) {
    int i = blockIdx.x * blockDim.x + threadIdx.x;
    if (i >= M) return;
    int s = (i < E) ? esrc[i] : (i - E);
    int d = (i < E) ? edst[i] : (i - E);
    int r = mapping[s], c = mapping[d];
    bool keep = (r >= 0) && (c >= 0);
    out_e[i]     = (float)(keep ? r : -1);
    out_e[M + i] = (float)(keep ? c : -1);
}

extern "C" void kernel_launch(void* const* d_in, const int* in_sizes, int n_in,
                              void* d_out, int out_size, void* d_ws, size_t ws_size,
                              hipStream_t stream) {
    const int C = 128;
    const float* x     = (const float*)d_in[0];
    const int*   eidx  = (const int*)d_in[1];
    const float* w_att = (const float*)d_in[2];
    const float* b_att = (const float*)d_in[3];
    const float* w1    = (const float*)d_in[4];
    const float* b1    = (const float*)d_in[5];
    const float* w2    = (const float*)d_in[6];
    const float* w3    = (const float*)d_in[7];
    const float* b3    = (const float*)d_in[8];

    const int N = in_sizes[0] / C;
    const int E = in_sizes[1] / 2;
    const int M = E + N;                // edges incl. self-loops
    const int K = (N + 1) / 2;          // ceil(0.5*N)
    int PAD = 1; while (PAD < N) PAD <<= 1;

    const int* esrc = eidx;
    const int* edst = eidx + E;

    // workspace carve-out (256B aligned slices)
    char* wsb = (char*)d_ws;
    size_t off = 0;
    auto alloc = [&](size_t bytes) -> void* {
        off = (off + 255) & ~(size_t)255;
        void* p = wsb + off;
        off += bytes;
        return p;
    };
    float*              p_src   = (float*)alloc((size_t)N * 4);
    float*              p_dst   = (float*)alloc((size_t)N * 4);
    float*              score   = (float*)alloc((size_t)M * 4);
    float*              exbuf   = (float*)alloc((size_t)M * 4);
    unsigned int*       menc    = (unsigned int*)alloc((size_t)N * 4);
    float*              den     = (float*)alloc((size_t)N * 4);
    float*              ssum    = (float*)alloc((size_t)N * 4);
    float*              aggv    = (float*)alloc((size_t)N * 4);
    float*              avec    = (float*)alloc((size_t)N * 4);
    float*              bvec    = (float*)alloc((size_t)N * 4);
    float*              xw3     = (float*)alloc((size_t)N * 4);
    float*              xagg    = (float*)alloc((size_t)N * C * 4);
    unsigned long long* keys    = (unsigned long long*)alloc((size_t)PAD * 8);
    int*                mapping = (int*)alloc((size_t)N * 4);

    // output layout: x_out[K*C] | edge_index[2*M] | batch[K] | perm[K]
    float* out_x     = (float*)d_out;
    float* out_e     = out_x + (size_t)K * C;
    float* out_batch = out_e + 2 * (size_t)M;
    float* out_perm  = out_batch + K;

    k_init<<<2048, 256, 0, stream>>>(xagg, menc, den, ssum, aggv, mapping, N, C);

    int tiles = (N + 15) / 16;
    int gemv_blocks = (tiles + 3) / 4;  // 4 waves (128 thr) per block
    k_gemv_att<<<gemv_blocks, 128, 0, stream>>>(x, w_att, p_src, p_dst, N, C);

    k_score<<<(M + 255) / 256, 256, 0, stream>>>(esrc, edst, p_src, p_dst, b_att,
                                                 score, menc, E, M);
    k_exp<<<(M + 255) / 256, 256, 0, stream>>>(edst, score, menc, exbuf, den, ssum, E, M);

    long aggT = (long)M * 32;
    k_agg<<<(int)((aggT + 255) / 256), 256, 0, stream>>>(esrc, edst, x, exbuf, den,
                                                         xagg, E, M, C);

    k_gemv_le<<<gemv_blocks, 128, 0, stream>>>(xagg, w1, w2, w3, b1, avec, bvec, xw3, N, C);
    k_leagg<<<(M + 255) / 256, 256, 0, stream>>>(esrc, edst, avec, bvec, aggv, E, M);

    k_fitness<<<(PAD + 255) / 256, 256, 0, stream>>>(aggv, xw3, b3, ssum, keys, N, PAD);

    // ---- bitonic sort (descending) with LDS-accelerated low-j passes ----
    if (PAD >= SORT_LOCAL) {
        int sblocks = PAD / SORT_LOCAL;
        k_bitonic_presort<<<sblocks, SORT_LOCAL / 2, 0, stream>>>(keys);
        for (int k2 = SORT_LOCAL << 1; k2 <= PAD; k2 <<= 1) {
            for (int j = k2 >> 1; j >= SORT_LOCAL; j >>= 1)
                k_bitonic<<<(PAD + 255) / 256, 256, 0, stream>>>(keys, j, k2, PAD);
            k_bitonic_merge_lds<<<sblocks, SORT_LOCAL / 2, 0, stream>>>(keys, k2);
        }
    } else {
        for (int k2 = 2; k2 <= PAD; k2 <<= 1)
            for (int j = k2 >> 1; j > 0; j >>= 1)
                k_bitonic<<<(PAD + 255) / 256, 256, 0, stream>>>(keys, j, k2, PAD);
    }

    k_map<<<(K + 255) / 256, 256, 0, stream>>>(keys, mapping, K);

    long outT = (long)K * 32;
    k_outx<<<(int)((outT + 255) / 256), 256, 0, stream>>>(keys, xagg, out_x, out_batch,
                                                          out_perm, K, C);
    k_edges<<<(M + 255) / 256, 256, 0, stream>>>(esrc, edst, mapping, out_e, E, M);
}